// RelationCorrelationModel_32976758898976
// MI455X (gfx1250) — compile-verified
//
#include <hip/hip_runtime.h>
#include <hip/hip_bf16.h>

// ---------------------------------------------------------------------------
// RelationCorrelationModel — algebraically collapsed implementation.
//
// Everything after the per-edge message construction is linear, so the whole
// model reduces to scalar messages:
//   u = W_fin[:128] @ W_score            (128)
//   v = W_fin[128:] @ W_score            (128)
//   c = b_fin @ W_score + b_score        (scalar)
//   WmatT[k][i] = (W_k @ u)[i]  (k: 0=tt,1=th,2=ht,3=hh; rows 4..15 zero pad)
//   Q = rel_emb @ Wmat                   [R,4]   <- fp32 WMMA 16x16x4 GEMM
//   bc[k] = b_k . u
//   T[r]  = rel_emb[r] . v
// per edge e: val = queries[e] ? 0 : sigmoid(corr[rt,rq]) * Q[rt][k] + bc[k]
//   scalar atomic add val & count to both endpoints
// per node:  s[n] = sum[n] / max(cnt[n],1)
// per query (deterministic compaction): out = s[h]+s[t]+T[rt]+c
// ---------------------------------------------------------------------------

typedef __attribute__((ext_vector_type(2))) float v2f;
typedef __attribute__((ext_vector_type(8))) float v8f;

#define EMBD 128
#define NNODES 200000   // reference constant N (num_nodes scalar lives on device)

// ---------------- utility: zero scratch ----------------
__global__ void k_zero_f32(float* __restrict__ p, long n) {
  long i = (long)blockIdx.x * blockDim.x + threadIdx.x;
  long stride = (long)gridDim.x * blockDim.x;
  for (; i < n; i += stride) p[i] = 0.0f;
}

// ---------------- stage A0: u, v, c ----------------
__global__ __launch_bounds__(256) void k_uvc(const float* __restrict__ W_fin,
                                             const float* __restrict__ b_fin,
                                             const float* __restrict__ W_score,
                                             const float* __restrict__ b_score,
                                             float* __restrict__ u,
                                             float* __restrict__ v,
                                             float* __restrict__ cvec) {
  int t = threadIdx.x;                 // 0..255 -> W_fin row t
  float acc = 0.f;
  #pragma unroll 8
  for (int j = 0; j < EMBD; ++j) acc += W_fin[t * EMBD + j] * W_score[j];
  if (t < EMBD) u[t] = acc; else v[t - EMBD] = acc;
  if (t == 0) {
    float cc = b_score[0];
    for (int j = 0; j < EMBD; ++j) cc += b_fin[j] * W_score[j];
    cvec[0] = cc;
  }
}

// ---------------- stage A1: WmatT, bc, T ----------------
// WmatT is [16][128]: row k (< 4) holds W_k @ u with K contiguous, so the
// WMMA B-matrix loads below are single b64 loads per lane. Rows 4..15 stay 0.
__global__ __launch_bounds__(256) void k_wmat_bc_T(
    const float* __restrict__ Wtt, const float* __restrict__ Wth,
    const float* __restrict__ Wht, const float* __restrict__ Whh,
    const float* __restrict__ btt, const float* __restrict__ bth,
    const float* __restrict__ bht, const float* __restrict__ bhh,
    const float* __restrict__ rel_emb,
    const float* __restrict__ u, const float* __restrict__ v,
    float* __restrict__ WmatT, float* __restrict__ bc, float* __restrict__ T,
    int R) {
  int tid = blockIdx.x * blockDim.x + threadIdx.x;
  if (tid < 4 * EMBD) {                         // WmatT[k*128+i] = sum_j W_k[i,j]*u[j]
    int i = tid >> 2, k = tid & 3;
    const float* Wp = (k == 0) ? Wtt : (k == 1) ? Wth : (k == 2) ? Wht : Whh;
    float acc = 0.f;
    #pragma unroll 8
    for (int j = 0; j < EMBD; ++j) acc += Wp[i * EMBD + j] * u[j];
    WmatT[k * EMBD + i] = acc;
  } else if (tid < 4 * EMBD + 4) {              // bc[k] = b_k . u
    int k = tid - 4 * EMBD;
    const float* bp = (k == 0) ? btt : (k == 1) ? bth : (k == 2) ? bht : bhh;
    float acc = 0.f;
    for (int j = 0; j < EMBD; ++j) acc += bp[j] * u[j];
    bc[k] = acc;
  } else if (tid < 4 * EMBD + 4 + R) {          // T[r] = rel_emb[r] . v
    int r = tid - (4 * EMBD + 4);
    float acc = 0.f;
    #pragma unroll 8
    for (int j = 0; j < EMBD; ++j) acc += rel_emb[r * EMBD + j] * v[j];
    T[r] = acc;
  }
}

// ---------------- stage A2: Q = rel_emb @ Wmat via fp32 WMMA ----------------
// One wave (32 lanes) computes one 16x16 output tile over K=128 with
// V_WMMA_F32_16X16X4_F32 (32 k-steps). ISA f32 layouts (CDNA5 ISA §7.12.2):
//   A 16x4:  lanes 0-15 M=0..15 {VGPR0=K0, VGPR1=K1}; lanes 16-31 {K2, K3}
//   B 4x16:  lanes hold N=lane%16; VGPR0/1 = K{0,1} (lo half) / K{2,3} (hi)
//   C/D:     VGPR j -> M=j (lanes 0-15), M=j+8 (lanes 16-31); N = lane%16
// Out-of-range A rows are clamped to row R-1 (in-bounds loads). No masking is
// needed: D[M][:] depends only on A row M, and rows >= R are discarded at the
// guarded store — so the k-loop is pure b64-load + v_wmma.
__global__ __launch_bounds__(32) void k_wmma_Q(const float* __restrict__ rel_emb,
                                               const float* __restrict__ WmatT,
                                               float* __restrict__ Q, int R) {
  const int lane = threadIdx.x;
  const int half = lane >> 4;
  const int n    = lane & 15;
  const int rowA = blockIdx.x * 16 + n;
  const int rowC = (rowA < R) ? rowA : (R - 1);
  const float* __restrict__ aptr = rel_emb + (long)rowC * EMBD + 2 * half;
  const float* __restrict__ bptr = WmatT + n * EMBD + 2 * half;
  v8f acc = {0.f, 0.f, 0.f, 0.f, 0.f, 0.f, 0.f, 0.f};
  #pragma unroll
  for (int kk = 0; kk < EMBD / 4; ++kk) {
    v2f a = *(const v2f*)(aptr + kk * 4);   // b64: A[M=row][K0..K1 (+2*half)]
    v2f b = *(const v2f*)(bptr + kk * 4);   // b64: B[K0..K1 (+2*half)][N=n]
    acc = __builtin_amdgcn_wmma_f32_16x16x4_f32(false, a, false, b,
                                                (short)0, acc, false, false);
  }
  #pragma unroll
  for (int j = 0; j < 8; ++j) {
    int row = blockIdx.x * 16 + j + 8 * half;
    if (row < R && n < 4) Q[row * 4 + n] = acc[j];
  }
}

// ---------------- stage B: per-edge scalar scatter ----------------
__global__ __launch_bounds__(256) void k_edges(
    const int2* __restrict__ ht, const int* __restrict__ rq,
    const int* __restrict__ rt, const int* __restrict__ rrel,
    const int* __restrict__ hot, const unsigned char* __restrict__ qry,
    const float* __restrict__ corr, const float* __restrict__ Q,
    const float* __restrict__ bc, float* __restrict__ sum,
    float* __restrict__ cnt, int M, int R) {
  long e = (long)blockIdx.x * blockDim.x + threadIdx.x;
  if (e >= M) return;
  int2 pair = ht[e];                         // .x = head, .y = tail
  float val = 0.f;
  if (!qry[e]) {
    int r = rt[e];
    float x = corr[(long)r * R + rq[e]];
    float g = 1.0f / (1.0f + __expf(-x));
    int k = ((rrel[e] == 0) ? 0 : 2) + ((hot[e] == 0) ? 0 : 1); // tt,th,ht,hh
    val = g * Q[r * 4 + k] + bc[k];
  }
  atomicAdd(&sum[pair.y], val);
  atomicAdd(&sum[pair.x], val);
  atomicAdd(&cnt[pair.y], 1.0f);
  atomicAdd(&cnt[pair.x], 1.0f);
}

// ---------------- stage C: node normalize (in place) ----------------
__global__ void k_nodes(float* __restrict__ sum, const float* __restrict__ cnt,
                        int N) {
  long n = (long)blockIdx.x * blockDim.x + threadIdx.x;
  if (n < N) sum[n] = sum[n] / fmaxf(cnt[n], 1.0f);
}

// ---------------- stage D: deterministic query compaction ----------------
__global__ __launch_bounds__(256) void k_bcount(const unsigned char* __restrict__ qry,
                                                unsigned int* __restrict__ bcnt,
                                                int M) {
  __shared__ int sh[256];
  int t = threadIdx.x;
  long e = (long)blockIdx.x * 256 + t;
  sh[t] = (e < M && qry[e]) ? 1 : 0;
  __syncthreads();
  for (int off = 128; off > 0; off >>= 1) {
    if (t < off) sh[t] += sh[t + off];
    __syncthreads();
  }
  if (t == 0) bcnt[blockIdx.x] = (unsigned int)sh[0];
}

__global__ void k_scan(const unsigned int* __restrict__ bcnt,
                       unsigned int* __restrict__ boff, int nb) {
  unsigned int run = 0;
  for (int i = 0; i < nb; ++i) { boff[i] = run; run += bcnt[i]; }
}

__global__ __launch_bounds__(256) void k_qout(
    const int2* __restrict__ ht, const int* __restrict__ rt,
    const unsigned char* __restrict__ qry, const float* __restrict__ s,
    const float* __restrict__ T, const float* __restrict__ cvec,
    const unsigned int* __restrict__ boff, float* __restrict__ out, int M) {
  __shared__ int sflag[256];
  int t = threadIdx.x;
  long e = (long)blockIdx.x * 256 + t;
  int f = (e < M && qry[e]) ? 1 : 0;
  sflag[t] = f;
  __syncthreads();
  for (int off = 1; off < 256; off <<= 1) {       // Hillis-Steele inclusive scan
    int vv = (t >= off) ? sflag[t - off] : 0;
    __syncthreads();
    sflag[t] += vv;
    __syncthreads();
  }
  if (f) {
    int pos = (int)boff[blockIdx.x] + sflag[t] - 1;
    int2 pair = ht[e];
    out[pos] = s[pair.x] + s[pair.y] + T[rt[e]] + cvec[0];
  }
}

// ---------------------------------------------------------------------------
extern "C" void kernel_launch(void* const* d_in, const int* in_sizes, int n_in,
                              void* d_out, int out_size, void* d_ws, size_t ws_size,
                              hipStream_t stream) {
  const int2* ht             = (const int2*)d_in[0];
  const int* r_q             = (const int*)d_in[1];
  const int* r_tensor        = (const int*)d_in[2];
  const int* r_relative      = (const int*)d_in[3];
  const int* h_or_t          = (const int*)d_in[4];
  const unsigned char* qry   = (const unsigned char*)d_in[5];
  // d_in[6] = num_nodes scalar (device); reference fixes N = 200000
  const float* rel_emb       = (const float*)d_in[7];
  const float* corr_table    = (const float*)d_in[8];
  const float* W_hh = (const float*)d_in[9];   const float* b_hh = (const float*)d_in[10];
  const float* W_ht = (const float*)d_in[11];  const float* b_ht = (const float*)d_in[12];
  const float* W_th = (const float*)d_in[13];  const float* b_th = (const float*)d_in[14];
  const float* W_tt = (const float*)d_in[15];  const float* b_tt = (const float*)d_in[16];
  const float* W_fin  = (const float*)d_in[17];
  const float* b_fin  = (const float*)d_in[18];
  const float* W_score = (const float*)d_in[19];
  const float* b_score = (const float*)d_in[20];
  float* out = (float*)d_out;

  const int M = in_sizes[0] / 2;
  const int R = in_sizes[7] / EMBD;
  const int N = NNODES;

  // ---- workspace layout (floats) ----
  float* ws    = (float*)d_ws;
  float* sum   = ws;                 // [N]
  float* cnt   = ws + N;             // [N]
  float* WmatT = ws + 2L * N;        // [16*128]  (rows 4..15 zero)
  float* Q     = WmatT + 2048;       // [512*4]
  float* u     = Q + 2048;           // [128]
  float* v     = u + 128;            // [128]
  float* bc    = v + 128;            // [16]
  float* cvec  = bc + 16;            // [16]
  float* T     = cvec + 16;          // [512]
  unsigned int* bcnt = (unsigned int*)(T + 512);   // [4096]
  unsigned int* boff = bcnt + 4096;                // [4096]

  const int nb = (M + 255) / 256;

  // zero sum/cnt/WmatT every call (determinism + WmatT pad rows)
  long zn = 2L * N + 2048;
  k_zero_f32<<<(int)((zn + 255) / 256), 256, 0, stream>>>(ws, zn);

  // A0: u, v, c
  k_uvc<<<1, 256, 0, stream>>>(W_fin, b_fin, W_score, b_score, u, v, cvec);

  // A1: WmatT (k order: tt, th, ht, hh), bc, T
  int a1_threads = 4 * EMBD + 4 + R;
  k_wmat_bc_T<<<(a1_threads + 255) / 256, 256, 0, stream>>>(
      W_tt, W_th, W_ht, W_hh, b_tt, b_th, b_ht, b_hh,
      rel_emb, u, v, WmatT, bc, T, R);

  // A2: Q = rel_emb @ Wmat  (fp32 WMMA, one wave per 16-row tile)
  k_wmma_Q<<<(R + 15) / 16, 32, 0, stream>>>(rel_emb, WmatT, Q, R);

  // B: scalar edge scatter
  k_edges<<<nb, 256, 0, stream>>>(ht, r_q, r_tensor, r_relative, h_or_t,
                                  qry, corr_table, Q, bc, sum, cnt, M, R);

  // C: normalize
  k_nodes<<<(N + 255) / 256, 256, 0, stream>>>(sum, cnt, N);

  // D: deterministic compaction of query edges -> output
  k_bcount<<<nb, 256, 0, stream>>>(qry, bcnt, M);
  k_scan<<<1, 1, 0, stream>>>(bcnt, boff, nb);
  k_qout<<<nb, 256, 0, stream>>>(ht, r_tensor, qry, sum, T, cvec, boff, out, M);

  (void)n_in; (void)out_size; (void)ws_size;
}